// TripletLoss_86071144612368
// MI455X (gfx1250) — compile-verified
//
#include <hip/hip_runtime.h>
#include <hip/hip_bf16.h>
#include <stdint.h>

// Problem constants (match reference)
#define B_ROWS   8192      // K * NUM
#define D_FEAT   4096
#define MARGIN_F 1.0f
#define EPS_F    1e-6f
#define THREADS  256       // 8 waves of 32

typedef __attribute__((ext_vector_type(2))) float v2f;
typedef __attribute__((ext_vector_type(8))) float v8f;

// ---------------------------------------------------------------------------
// 32-lane wave sum using V_WMMA_F32_16X16X4_F32 against a ones B-matrix.
// A layout (16x4 f32): lanes 0-15 hold A[m=lane][K0,K1] in vgpr0/1,
// lanes 16-31 hold A[m=lane-16][K2,K3]. With A = {v, 0} per lane and B = 1,
// D[m][n] = v(lane m) + v(lane m+16). C layout: vgpr j, lanes 0-15 => M=j,
// lanes 16-31 => M=j+8. So sum(c[0..7]) gives half-sums; one xor-16 shuffle
// completes the full 32-lane sum in every lane. All adds are exact f32.
// ---------------------------------------------------------------------------
__device__ __forceinline__ float wave_sum_wmma(float v) {
  v2f a = {v, 0.0f};
  v2f b = {1.0f, 1.0f};
  v8f c = {};
  c = __builtin_amdgcn_wmma_f32_16x16x4_f32(
      /*neg_a=*/false, a, /*neg_b=*/false, b,
      /*c_mod=*/(short)0, c, /*reuse_a=*/false, /*reuse_b=*/false);
  float s = c[0] + c[1] + c[2] + c[3] + c[4] + c[5] + c[6] + c[7];
  return s + __shfl_xor(s, 16, 32);
}

__device__ __forceinline__ void acc_pair(float a, float b, float& s1, float& s2) {
  float d = a - b;
  s1 += fabsf(d);
  s2 = fmaf(d, d, s2);
}

// ---------------------------------------------------------------------------
// One workgroup per row. Stage anchor/pos/neg rows (3 x 16 KB) into LDS with
// GLOBAL_LOAD_ASYNC_TO_LDS_B128 (ASYNCcnt path), then accumulate
// sum|d| and sum d^2 for both pairs, reduce with WMMA + LDS, write per-row
// loss and accuracy flag to workspace.
// ---------------------------------------------------------------------------
__global__ __launch_bounds__(THREADS) void triplet_rows_kernel(
    const float* __restrict__ feat, const int* __restrict__ pos_idx,
    const int* __restrict__ neg_idx, float* __restrict__ ws) {
  __shared__ float sm[3 * D_FEAT];   // [a | p | n], 48 KB
  __shared__ float sred[8 * 4];      // per-wave partials

  const int row  = blockIdx.x;
  const int t    = threadIdx.x;
  const int lane = t & 31;
  const int wv   = t >> 5;

  const float* arow = feat + (size_t)row * D_FEAT;
  const float* prow = feat + (size_t)pos_idx[row] * D_FEAT;
  const float* nrow = feat + (size_t)neg_idx[row] * D_FEAT;

  // Async-stage: thread t owns float4 chunks {t, t+256, t+512, t+768} of each
  // row and later consumes exactly those chunks, so per-wave
  // s_wait_asynccnt 0 is sufficient (no cross-wave LDS consumption).
#pragma unroll
  for (int c = 0; c < 4; ++c) {
    const int i = (t + THREADS * c) * 4;          // float index of 16B chunk
    uint32_t l0 = (uint32_t)(uintptr_t)(&sm[0 * D_FEAT + i]);
    uint32_t l1 = (uint32_t)(uintptr_t)(&sm[1 * D_FEAT + i]);
    uint32_t l2 = (uint32_t)(uintptr_t)(&sm[2 * D_FEAT + i]);
    uint64_t g0 = (uint64_t)(uintptr_t)(arow + i);
    uint64_t g1 = (uint64_t)(uintptr_t)(prow + i);
    uint64_t g2 = (uint64_t)(uintptr_t)(nrow + i);
    asm volatile("global_load_async_to_lds_b128 %0, %1, off"
                 :: "v"(l0), "v"(g0) : "memory");
    asm volatile("global_load_async_to_lds_b128 %0, %1, off"
                 :: "v"(l1), "v"(g1) : "memory");
    asm volatile("global_load_async_to_lds_b128 %0, %1, off"
                 :: "v"(l2), "v"(g2) : "memory");
  }
  asm volatile("s_wait_asynccnt 0" ::: "memory");

  float s1p = 0.0f, s2p = 0.0f, s1n = 0.0f, s2n = 0.0f;
#pragma unroll
  for (int c = 0; c < 4; ++c) {
    const int i = (t + THREADS * c) * 4;
    float4 a4 = *(const float4*)(&sm[0 * D_FEAT + i]);
    float4 p4 = *(const float4*)(&sm[1 * D_FEAT + i]);
    float4 n4 = *(const float4*)(&sm[2 * D_FEAT + i]);
    acc_pair(a4.x, p4.x, s1p, s2p); acc_pair(a4.x, n4.x, s1n, s2n);
    acc_pair(a4.y, p4.y, s1p, s2p); acc_pair(a4.y, n4.y, s1n, s2n);
    acc_pair(a4.z, p4.z, s1p, s2p); acc_pair(a4.z, n4.z, s1n, s2n);
    acc_pair(a4.w, p4.w, s1p, s2p); acc_pair(a4.w, n4.w, s1n, s2n);
  }

  // Wave-level reduction via WMMA (EXEC is all-1s here: full block, no
  // divergence), then cross-wave via LDS.
  float w1p = wave_sum_wmma(s1p);
  float w2p = wave_sum_wmma(s2p);
  float w1n = wave_sum_wmma(s1n);
  float w2n = wave_sum_wmma(s2n);

  if (lane == 0) {
    sred[wv * 4 + 0] = w1p;
    sred[wv * 4 + 1] = w2p;
    sred[wv * 4 + 2] = w1n;
    sred[wv * 4 + 3] = w2n;
  }
  __syncthreads();

  if (t == 0) {
    float S1p = 0.0f, S2p = 0.0f, S1n = 0.0f, S2n = 0.0f;
    for (int w = 0; w < 8; ++w) {     // fixed order: deterministic
      S1p += sred[w * 4 + 0];
      S2p += sred[w * 4 + 1];
      S1n += sred[w * 4 + 2];
      S2n += sred[w * 4 + 3];
    }
    // sum((|d|+eps)^2) = sum d^2 + 2*eps*sum|d| + D*eps^2  (exact rewrite)
    const float twoEps = 2.0f * EPS_F;
    const float epsTerm = (float)D_FEAT * EPS_F * EPS_F;
    float dp = sqrtf(S2p + twoEps * S1p + epsTerm);
    float dn = sqrtf(S2n + twoEps * S1n + epsTerm);
    float loss = fmaxf(dp - dn + MARGIN_F, 0.0f);
    float accf = (sqrtf(S2p) < sqrtf(S2n)) ? 1.0f : 0.0f;
    ws[row]          = loss;
    ws[B_ROWS + row] = accf;
  }
}

// ---------------------------------------------------------------------------
// Deterministic final reduction: single block, fixed summation order.
// d_out[0] = acc mean, d_out[1] = loss mean (reference tuple order).
// ---------------------------------------------------------------------------
__global__ __launch_bounds__(THREADS) void triplet_final_kernel(
    const float* __restrict__ ws, float* __restrict__ out) {
  __shared__ float sl[THREADS];
  __shared__ float sa[THREADS];
  const int t = threadIdx.x;
  float l = 0.0f, a = 0.0f;
  for (int k = 0; k < B_ROWS / THREADS; ++k) {   // fixed order per thread
    l += ws[t + THREADS * k];
    a += ws[B_ROWS + t + THREADS * k];
  }
  sl[t] = l;
  sa[t] = a;
  __syncthreads();
  for (int s = THREADS / 2; s > 0; s >>= 1) {    // fixed pairing: deterministic
    if (t < s) {
      sl[t] += sl[t + s];
      sa[t] += sa[t + s];
    }
    __syncthreads();
  }
  if (t == 0) {
    out[0] = sa[0] / (float)B_ROWS;  // acc
    out[1] = sl[0] / (float)B_ROWS;  // loss
  }
}

extern "C" void kernel_launch(void* const* d_in, const int* in_sizes, int n_in,
                              void* d_out, int out_size, void* d_ws, size_t ws_size,
                              hipStream_t stream) {
  const float* feat = (const float*)d_in[0];
  const int*   pos  = (const int*)d_in[1];
  const int*   neg  = (const int*)d_in[2];
  float* ws  = (float*)d_ws;    // needs 2*B_ROWS floats = 64 KB
  float* out = (float*)d_out;   // [acc, loss]

  triplet_rows_kernel<<<B_ROWS, THREADS, 0, stream>>>(feat, pos, neg, ws);
  triplet_final_kernel<<<1, THREADS, 0, stream>>>(ws, out);
}